// MultiHeadAttention_12953621365021
// MI455X (gfx1250) — compile-verified
//
#include <hip/hip_runtime.h>
#include <math.h>

typedef float v2f __attribute__((ext_vector_type(2)));
typedef float v8f __attribute__((ext_vector_type(8)));

#define Bq   2
#define Sq   2048
#define Hq   1024
#define NHq  16
#define HDq  64
#define Mtot (Bq * Sq)   // 4096

static __device__ __forceinline__ v8f wmma_f32(v2f a, v2f b, v8f c) {
    // V_WMMA_F32_16X16X4_F32 : D(16x16 f32) = A(16x4 f32) * B(4x16 f32) + C
    return __builtin_amdgcn_wmma_f32_16x16x4_f32(false, a, false, b, (short)0, c,
                                                 false, false);
}

// ---------------------------------------------------------------------------
// GEMM: Out = X[M,K] @ W[N,K]^T + bias  (M=4096, N=K=1024)
// MODE 0: scatter to head layout [B,NH,S,HD]
// MODE 1: Out[m*H+n] = acc + bias[n] + res[m*H+n]   (row-major + residual)
// Block = 128 thr (4 waves); block tile 64x128; wave tile 32x64 (2x4 WMMA
// tiles) -> 6 global_load_b64 per 8 WMMAs per K-step of 4.
// ---------------------------------------------------------------------------
template <int MODE>
__global__ __launch_bounds__(128) void gemm_wmma_kernel(
    const float* __restrict__ X, const float* __restrict__ W,
    const float* __restrict__ bias, const float* __restrict__ res,
    float* __restrict__ out) {
    const int lane = threadIdx.x & 31;
    const int wv   = threadIdx.x >> 5;
    const int lm   = lane & 15;
    const int hl   = lane >> 4;
    const int mBase = blockIdx.y * 64 + (wv >> 1) * 32;
    const int nBase = blockIdx.x * 128 + (wv & 1) * 64;
    const int ko    = 2 * hl;

    const float* x0 = X + (size_t)(mBase + lm) * Hq + ko;
    const float* x1 = X + (size_t)(mBase + 16 + lm) * Hq + ko;
    const float* w0 = W + (size_t)(nBase + lm) * Hq + ko;
    const float* w1 = W + (size_t)(nBase + 16 + lm) * Hq + ko;
    const float* w2 = W + (size_t)(nBase + 32 + lm) * Hq + ko;
    const float* w3 = W + (size_t)(nBase + 48 + lm) * Hq + ko;

    v8f acc[2][4];
#pragma unroll
    for (int i = 0; i < 2; ++i)
#pragma unroll
        for (int j = 0; j < 4; ++j)
#pragma unroll
            for (int r = 0; r < 8; ++r) acc[i][j][r] = 0.f;

#pragma unroll 2
    for (int k0 = 0; k0 < Hq; k0 += 4) {
        v2f a0 = *(const v2f*)(x0 + k0);
        v2f a1 = *(const v2f*)(x1 + k0);
        v2f b0 = *(const v2f*)(w0 + k0);
        v2f b1 = *(const v2f*)(w1 + k0);
        v2f b2 = *(const v2f*)(w2 + k0);
        v2f b3 = *(const v2f*)(w3 + k0);
        acc[0][0] = wmma_f32(a0, b0, acc[0][0]);
        acc[0][1] = wmma_f32(a0, b1, acc[0][1]);
        acc[0][2] = wmma_f32(a0, b2, acc[0][2]);
        acc[0][3] = wmma_f32(a0, b3, acc[0][3]);
        acc[1][0] = wmma_f32(a1, b0, acc[1][0]);
        acc[1][1] = wmma_f32(a1, b1, acc[1][1]);
        acc[1][2] = wmma_f32(a1, b2, acc[1][2]);
        acc[1][3] = wmma_f32(a1, b3, acc[1][3]);
    }

#pragma unroll
    for (int ii = 0; ii < 2; ++ii) {
#pragma unroll
        for (int jj = 0; jj < 4; ++jj) {
            const int   n  = nBase + jj * 16 + lm;
            const float bv = bias[n];
#pragma unroll
            for (int r = 0; r < 8; ++r) {
                const int m = mBase + ii * 16 + r + 8 * hl;
                float v = acc[ii][jj][r] + bv;
                if (MODE == 0) {
                    const int b = m >> 11, s = m & (Sq - 1);
                    const int h = n >> 6, d = n & (HDq - 1);
                    out[(size_t)((b * NHq + h) * Sq + s) * HDq + d] = v;
                } else {
                    out[(size_t)m * Hq + n] = v + res[(size_t)m * Hq + n];
                }
            }
        }
    }
}

// ---------------------------------------------------------------------------
// Flash attention (causal). Grid: (S/64 qtiles, B*NH). Block = 128 (4 waves).
// Each wave owns 16 queries; K loop over 16-key tiles with online softmax.
// Q@K^T and P@V both via V_WMMA_F32_16X16X4_F32.
// ---------------------------------------------------------------------------
__global__ __launch_bounds__(128) void attn_kernel(const float* __restrict__ Q,
                                                   const float* __restrict__ K,
                                                   const float* __restrict__ V,
                                                   float* __restrict__ ctx) {
    __shared__ float lds_p[4][16 * 18];  // per-wave P tile, padded stride 18

    const int lane = threadIdx.x & 31;
    const int wv   = threadIdx.x >> 5;
    const int lm   = lane & 15;
    const int hl   = lane >> 4;
    const int qt   = blockIdx.x;
    const int bh   = blockIdx.y;

    const float* Qb = Q + (size_t)bh * Sq * HDq;
    const float* Kb = K + (size_t)bh * Sq * HDq;
    const float* Vb = V + (size_t)bh * Sq * HDq;

    const int qbase = qt * 64 + wv * 16;

    // Preload Q fragments: 16 k-steps over HD=64
    v2f qa[16];
    const float* qrow = Qb + (size_t)(qbase + lm) * HDq + 2 * hl;
#pragma unroll
    for (int kk = 0; kk < 16; ++kk) qa[kk] = *(const v2f*)(qrow + kk * 4);

    float m_run[8], l_run[8];
    v8f   oacc[4];
#pragma unroll
    for (int r = 0; r < 8; ++r) { m_run[r] = -1e30f; l_run[r] = 0.f; }
#pragma unroll
    for (int j = 0; j < 4; ++j)
#pragma unroll
        for (int r = 0; r < 8; ++r) oacc[j][r] = 0.f;

    float* pw = &lds_p[wv][0];
    const int ntiles = qt * 4 + 4;  // uniform per block (causal upper bound)

    for (int kt = 0; kt < ntiles; ++kt) {
        const int kbase = kt * 16;

        // S tile = Q(16x64) @ K^T(64x16)
        v8f s;
#pragma unroll
        for (int r = 0; r < 8; ++r) s[r] = 0.f;
        const float* krow = Kb + (size_t)(kbase + lm) * HDq + 2 * hl;
#pragma unroll
        for (int kk = 0; kk < 16; ++kk) {
            v2f kb = *(const v2f*)(krow + kk * 4);
            s      = wmma_f32(qa[kk], kb, s);
        }

        // scale + causal mask + online softmax update (rows = r + 8*hl)
        const int kcol = kbase + lm;
#pragma unroll
        for (int r = 0; r < 8; ++r) {
            const int qg = qbase + r + 8 * hl;
            float sv     = s[r] * 0.125f;  // 1/sqrt(64)
            sv           = (kcol <= qg) ? sv : -1e30f;
            float tm     = sv;
            tm = fmaxf(tm, __shfl_xor(tm, 1, 16));
            tm = fmaxf(tm, __shfl_xor(tm, 2, 16));
            tm = fmaxf(tm, __shfl_xor(tm, 4, 16));
            tm = fmaxf(tm, __shfl_xor(tm, 8, 16));
            const float nm    = fmaxf(m_run[r], tm);
            const float alpha = __expf(m_run[r] - nm);
            m_run[r]          = nm;
            const float pv    = __expf(sv - nm);
            float rs          = pv;
            rs += __shfl_xor(rs, 1, 16);
            rs += __shfl_xor(rs, 2, 16);
            rs += __shfl_xor(rs, 4, 16);
            rs += __shfl_xor(rs, 8, 16);
            l_run[r] = l_run[r] * alpha + rs;
#pragma unroll
            for (int j = 0; j < 4; ++j) oacc[j][r] *= alpha;
            pw[(r + 8 * hl) * 18 + lm] = pv;  // C-layout -> LDS
        }
        __syncthreads();  // LDS P visible (re-layout for A operand)

        // ctx += P(16x16) @ V(16x64)
#pragma unroll
        for (int kk2 = 0; kk2 < 4; ++kk2) {
            v2f pa = *(const v2f*)(pw + lm * 18 + kk2 * 4 + 2 * hl);
            const float* vrow = Vb + (size_t)(kbase + kk2 * 4 + 2 * hl) * HDq;
#pragma unroll
            for (int j = 0; j < 4; ++j) {
                v2f bv;
                bv[0]   = vrow[j * 16 + lm];
                bv[1]   = vrow[HDq + j * 16 + lm];
                oacc[j] = wmma_f32(pa, bv, oacc[j]);
            }
        }
        __syncthreads();
    }

    // normalize and store ctx as [B,S,H]
    const int b = bh >> 4, h = bh & 15;
#pragma unroll
    for (int j = 0; j < 4; ++j) {
#pragma unroll
        for (int r = 0; r < 8; ++r) {
            const int qg = qbase + r + 8 * hl;
            const int d  = j * 16 + lm;
            ctx[(size_t)(b * Sq + qg) * Hq + h * HDq + d] = oacc[j][r] / l_run[r];
        }
    }
}

// ---------------------------------------------------------------------------
// In-place LayerNorm over H=1024 per row. Grid = B*S rows, 256 threads.
// ---------------------------------------------------------------------------
__global__ __launch_bounds__(256) void ln_kernel(float* __restrict__ y,
                                                 const float* __restrict__ g,
                                                 const float* __restrict__ bta) {
    __shared__ float smem[16];
    const int row = blockIdx.x;
    float* yr     = y + (size_t)row * Hq;

    float4 v = ((const float4*)yr)[threadIdx.x];
    float  s = v.x + v.y + v.z + v.w;
    float  q = v.x * v.x + v.y * v.y + v.z * v.z + v.w * v.w;
#pragma unroll
    for (int off = 16; off > 0; off >>= 1) {
        s += __shfl_xor(s, off, 32);
        q += __shfl_xor(q, off, 32);
    }
    const int wv = threadIdx.x >> 5;
    const int ln = threadIdx.x & 31;
    if (ln == 0) { smem[wv] = s; smem[8 + wv] = q; }
    __syncthreads();
    if (wv == 0) {
        float s2 = (ln < 8) ? smem[ln] : 0.f;
        float q2 = (ln < 8) ? smem[8 + ln] : 0.f;
#pragma unroll
        for (int off = 4; off > 0; off >>= 1) {
            s2 += __shfl_xor(s2, off, 8);
            q2 += __shfl_xor(q2, off, 8);
        }
        if (ln == 0) { smem[0] = s2; smem[8] = q2; }
    }
    __syncthreads();
    const float mean = smem[0] * (1.f / Hq);
    const float var  = smem[8] * (1.f / Hq) - mean * mean;
    const float rstd = rsqrtf(var + 1e-12f);

    float4 gg = ((const float4*)g)[threadIdx.x];
    float4 bb = ((const float4*)bta)[threadIdx.x];
    float4 o;
    o.x = (v.x - mean) * rstd * gg.x + bb.x;
    o.y = (v.y - mean) * rstd * gg.y + bb.y;
    o.z = (v.z - mean) * rstd * gg.z + bb.z;
    o.w = (v.w - mean) * rstd * gg.w + bb.w;
    ((float4*)yr)[threadIdx.x] = o;
}

// ---------------------------------------------------------------------------
extern "C" void kernel_launch(void* const* d_in, const int* in_sizes, int n_in,
                              void* d_out, int out_size, void* d_ws,
                              size_t ws_size, hipStream_t stream) {
    const float* x   = (const float*)d_in[0];
    const float* Wq_ = (const float*)d_in[1];
    const float* bq_ = (const float*)d_in[2];
    const float* Wk_ = (const float*)d_in[3];
    const float* bk_ = (const float*)d_in[4];
    const float* Wv_ = (const float*)d_in[5];
    const float* bv_ = (const float*)d_in[6];
    const float* Wo_ = (const float*)d_in[7];
    const float* bo_ = (const float*)d_in[8];
    const float* g_  = (const float*)d_in[9];
    const float* b_  = (const float*)d_in[10];
    float* out       = (float*)d_out;

    float* Qw = (float*)d_ws;                  // [B,NH,S,HD] 16 MB
    float* Kw = Qw + (size_t)Mtot * Hq;        // 16 MB
    float* Vw = Kw + (size_t)Mtot * Hq;        // 16 MB
    float* Cw = Vw + (size_t)Mtot * Hq;        // ctx [B,S,H] 16 MB

    dim3 gg(Hq / 128, Mtot / 64);
    dim3 gb(128);
    gemm_wmma_kernel<0><<<gg, gb, 0, stream>>>(x, Wq_, bq_, nullptr, Qw);
    gemm_wmma_kernel<0><<<gg, gb, 0, stream>>>(x, Wk_, bk_, nullptr, Kw);
    gemm_wmma_kernel<0><<<gg, gb, 0, stream>>>(x, Wv_, bv_, nullptr, Vw);
    attn_kernel<<<dim3(Sq / 64, Bq * NHq), dim3(128), 0, stream>>>(Qw, Kw, Vw, Cw);
    gemm_wmma_kernel<1><<<gg, gb, 0, stream>>>(Cw, Wo_, bo_, x, out);
    ln_kernel<<<dim3(Mtot), dim3(256), 0, stream>>>(out, g_, b_);
}